// dMAELoss_31954556682741
// MI455X (gfx1250) — compile-verified
//
#include <hip/hip_runtime.h>
#include <hip/hip_bf16.h>
#include <math.h>
#include <stdint.h>

typedef __attribute__((ext_vector_type(2))) float v2f;
typedef __attribute__((ext_vector_type(4))) float f32x4;
typedef __attribute__((ext_vector_type(8))) float v8f;
typedef __attribute__((ext_vector_type(4))) int   v4i;

#define LPTS   1024
#define NBATCH 32
#define EPSV   1e-4f
#define ZDIV   10.0f
#define ROWTILES (LPTS / 16)   // 64

#if defined(__has_builtin)
#  if __has_builtin(__builtin_amdgcn_global_load_async_to_lds_b128)
#    define HAVE_ASYNC_LDS 1
#  endif
#  if __has_builtin(__builtin_amdgcn_s_wait_asynccnt)
#    define HAVE_WAIT_ASYNC 1
#  endif
#endif

// One block = 8 waves; handles one 16-row strip of one batch's L x L matrix.
// Points are staged flat into LDS with async b128 copies, then repacked as
// {x,y,z,0} (16B stride) so each WMMA A/B fragment is one unconditional
// ds_load_b64 (K=3 padded to 4 comes from memory, no selects / exec masking).
// Each wave computes 8 16x16 tiles of BOTH Gram matrices with
// V_WMMA_F32_16X16X4_F32 and converts to distances via
// d^2 = |xi|^2 + |xj|^2 - 2 xi.xj.
__global__ __launch_bounds__(256) void dmae_tile_kernel(
    const float* __restrict__ pred,
    const float* __restrict__ gt,
    float* __restrict__ partial)
{
    __shared__ float sFP[3 * LPTS];  // pred xyz flat (staging)
    __shared__ float sFG[3 * LPTS];  // gt   xyz flat (staging)
    __shared__ float sP4[4 * LPTS];  // pred {x,y,z,0} padded
    __shared__ float sG4[4 * LPTS];  // gt   {x,y,z,0} padded
    __shared__ float sNP[LPTS];      // pred |x|^2
    __shared__ float sNG[LPTS];      // gt   |x|^2
    __shared__ float swave[8];

    const int b    = blockIdx.x / ROWTILES;   // batch
    const int it   = blockIdx.x % ROWTILES;   // row tile
    const int i0   = it * 16;
    const int tid  = threadIdx.x;
    const int lane = tid & 31;
    const int wave = tid >> 5;                // 0..7
    const int half = lane >> 4;               // 0 or 1
    const int lm   = lane & 15;

    const float* __restrict__ pb = pred + (size_t)b * LPTS * 3;
    const float* __restrict__ gb = gt   + (size_t)b * LPTS * 3;

    // ---- Stage 2 x 3072 floats (24 KB) into LDS: 3 x b128 per thread each ----
#if defined(HAVE_ASYNC_LDS)
#pragma unroll
    for (int k = 0; k < 3; ++k) {
        const int idx = (k * 256 + tid) * 4;  // float index, 16B aligned
        __builtin_amdgcn_global_load_async_to_lds_b128(
            (v4i*)(pb + idx), (v4i*)&sFP[idx], 0, 0);
        __builtin_amdgcn_global_load_async_to_lds_b128(
            (v4i*)(gb + idx), (v4i*)&sFG[idx], 0, 0);
    }
#  if defined(HAVE_WAIT_ASYNC)
    __builtin_amdgcn_s_wait_asynccnt(0);
#  else
    asm volatile("s_wait_asynccnt 0x0" ::: "memory");
#  endif
#else
#pragma unroll
    for (int k = 0; k < 3; ++k) {
        const int idx = (k * 256 + tid) * 4;
        *(f32x4*)&sFP[idx] = *(const f32x4*)&pb[idx];
        *(f32x4*)&sFG[idx] = *(const f32x4*)&gb[idx];
    }
#endif
    __syncthreads();

    // ---- Repack to 16B-stride {x,y,z,0} and precompute squared norms ----
#pragma unroll
    for (int i = tid; i < LPTS; i += 256) {
        float x = sFP[3 * i], y = sFP[3 * i + 1], z = sFP[3 * i + 2];
        f32x4 v; v.x = x; v.y = y; v.z = z; v.w = 0.0f;
        *(f32x4*)&sP4[4 * i] = v;
        sNP[i] = x * x + y * y + z * z;
        x = sFG[3 * i]; y = sFG[3 * i + 1]; z = sFG[3 * i + 2];
        f32x4 w; w.x = x; w.y = y; w.z = z; w.w = 0.0f;
        *(f32x4*)&sG4[4 * i] = w;
        sNG[i] = x * x + y * y + z * z;
    }
    __syncthreads();

    // ---- A tile: rows i0..i0+15, 16x4 f32 ----
    // ISA layout: v0 = K0 (lanes 0-15) / K2 (lanes 16-31); v1 = K1 / K3.
    // Padded layout -> one b64 load: lanes 0-15 {x,y}, lanes 16-31 {z,0}.
    const int arow = i0 + lm;
    const int koff = 2 * half;                // 0 or 2 floats
    const v2f aP = *(const v2f*)&sP4[arow * 4 + koff];
    const v2f aG = *(const v2f*)&sG4[arow * 4 + koff];

    // Row norms for this lane's C/D rows: M = r + 8*half
    float nrP[8], nrG[8];
#pragma unroll
    for (int r = 0; r < 8; ++r) {
        const int m = i0 + r + 8 * half;
        nrP[r] = sNP[m];
        nrG[r] = sNG[m];
    }

    float sum = 0.0f;

    for (int t = 0; t < 8; ++t) {
        const int j0   = (wave * 8 + t) * 16;  // column tile base
        const int brow = j0 + lm;

        // B tile = A^T of the 16 column points; same per-lane packing
        const v2f bP = *(const v2f*)&sP4[brow * 4 + koff];
        const v2f bG = *(const v2f*)&sG4[brow * 4 + koff];

        const float ncP = sNP[brow];
        const float ncG = sNG[brow];

        v8f cz = {};
        v8f dotP = __builtin_amdgcn_wmma_f32_16x16x4_f32(
            false, aP, false, bP, (short)0, cz, false, false);
        v8f dotG = __builtin_amdgcn_wmma_f32_16x16x4_f32(
            false, aG, false, bG, (short)0, cz, false, false);

        // Epilogue: distances + masked |diff| (raw v_sqrt_f32)
#pragma unroll
        for (int r = 0; r < 8; ++r) {
            const int row = i0 + r + 8 * half;
            const int col = brow;
            const float d2p = fmaxf(nrP[r] + ncP - 2.0f * dotP[r], 0.0f) + EPSV;
            const float d2g = fmaxf(nrG[r] + ncG - 2.0f * dotG[r], 0.0f) + EPSV;
            const float dp = __builtin_amdgcn_sqrtf(d2p);
            const float dg = __builtin_amdgcn_sqrtf(d2g);
            sum += (row != col) ? fabsf(dp - dg) : 0.0f;
        }
    }

    // wave32 shuffle reduce
#pragma unroll
    for (int off = 16; off > 0; off >>= 1)
        sum += __shfl_down(sum, off, 32);

    if (lane == 0) swave[wave] = sum;
    __syncthreads();
    if (tid == 0) {
        float s = 0.0f;
#pragma unroll
        for (int w = 0; w < 8; ++w) s += swave[w];
        partial[blockIdx.x] = s;
    }
}

// Deterministic final reduce of per-block partials + scaling.
__global__ __launch_bounds__(256) void dmae_reduce_kernel(
    const float* __restrict__ partial, int n, float* __restrict__ out)
{
    __shared__ float ws[256];
    float s = 0.0f;
    for (int i = threadIdx.x; i < n; i += 256) s += partial[i];
    ws[threadIdx.x] = s;
    __syncthreads();
    for (int off = 128; off > 0; off >>= 1) {
        if ((int)threadIdx.x < off) ws[threadIdx.x] += ws[threadIdx.x + off];
        __syncthreads();
    }
    if (threadIdx.x == 0) {
        const float den = (float)NBATCH * (float)LPTS * (float)(LPTS - 1) * ZDIV;
        out[0] = ws[0] / den;
    }
}

extern "C" void kernel_launch(void* const* d_in, const int* in_sizes, int n_in,
                              void* d_out, int out_size, void* d_ws, size_t ws_size,
                              hipStream_t stream)
{
    const float* pred = (const float*)d_in[0];
    const float* gt   = (const float*)d_in[1];
    float* out        = (float*)d_out;
    float* partial    = (float*)d_ws;   // NBATCH*ROWTILES floats = 8 KB

    const int nBlocks = NBATCH * ROWTILES;  // 2048
    dmae_tile_kernel<<<nBlocks, 256, 0, stream>>>(pred, gt, partial);
    dmae_reduce_kernel<<<1, 256, 0, stream>>>(partial, nBlocks, out);
}